// ConnectLoss_24154896073265
// MI455X (gfx1250) — compile-verified
//
#include <hip/hip_runtime.h>

// ConnectLoss: loss = sum( relu(|x[:,j]-x[:,j-1]| - 1) * mask ), column 0 term = 0.
// Memory-bound: 256 MB streamed, ~11 us at 23.3 TB/s on MI455X. Strategy:
//   kernel 1: each thread processes 16 contiguous elements per iteration
//             (4x global_load_b128 x, 4x non-temporal b128 mask, 1x unconditional
//             b32 predecessor via index-select -> no divergent branch),
//             per-thread f32 partial, wave32 reduction via V_WMMA_F32_16X16X4_F32
//             (B = ones), LDS combine of 8 wave sums, one partial per block -> d_ws.
//   kernel 2: single block, fixed-order tree reduction in double -> d_out[0].
// Fully deterministic (no atomics), graph-capture safe.

typedef float v2f __attribute__((ext_vector_type(2)));
typedef float v4f __attribute__((ext_vector_type(4)));
typedef float v8f __attribute__((ext_vector_type(8)));

#define NBLK 2048
#define NTHR 256
#define NWAVE (NTHR / 32)

__device__ __forceinline__ float quad_terms(float xp, v4f xv, v4f mv, float& acc) {
  const float d0 = xv[0] - xp;
  const float d1 = xv[1] - xv[0];
  const float d2 = xv[2] - xv[1];
  const float d3 = xv[3] - xv[2];
  acc += fmaxf(fabsf(d0) - 1.0f, 0.0f) * mv[0];
  acc += fmaxf(fabsf(d1) - 1.0f, 0.0f) * mv[1];
  acc += fmaxf(fabsf(d2) - 1.0f, 0.0f) * mv[2];
  acc += fmaxf(fabsf(d3) - 1.0f, 0.0f) * mv[3];
  return xv[3];
}

__global__ __launch_bounds__(NTHR) void connect_loss_partial(
    const float* __restrict__ x, const float* __restrict__ mask,
    float* __restrict__ partials, int nsegs /* 16-element segments */) {
  float acc = 0.0f;
  const int stride = gridDim.x * blockDim.x;
  for (int seg = blockIdx.x * blockDim.x + threadIdx.x; seg < nsegs; seg += stride) {
    const int f = seg << 4;          // flat element index of this 16-elem segment
    const int col = f & 511;         // LENGTH = 512

    const v4f* xg = reinterpret_cast<const v4f*>(x + f);
    const v4f* mg = reinterpret_cast<const v4f*>(mask + f);
    const v4f x0 = xg[0], x1 = xg[1], x2 = xg[2], x3 = xg[3];
    // mask is read exactly once across the whole kernel: non-temporal,
    // don't let it evict x lines from the 192 MB L2.
    const v4f m0 = __builtin_nontemporal_load(mg + 0);
    const v4f m1 = __builtin_nontemporal_load(mg + 1);
    const v4f m2 = __builtin_nontemporal_load(mg + 2);
    const v4f m3 = __builtin_nontemporal_load(mg + 3);

    // Predecessor of the segment, loaded unconditionally (no divergent branch):
    // col==0 -> pidx==f -> xp == x0[0] -> first diff exactly 0.
    const int pidx = f - (int)(col != 0);
    float xp = x[pidx];
    xp = quad_terms(xp, x0, m0, acc);
    xp = quad_terms(xp, x1, m1, acc);
    xp = quad_terms(xp, x2, m2, acc);
    (void)quad_terms(xp, x3, m3, acc);
  }

  // ---- wave32 reduction via WMMA (f32, exact) ----
  // A (16x4 f32, 2 VGPRs): a[0] -> A[m,0] (lanes 0-15) / A[m,2] (lanes 16-31); a[1] -> 0.
  // B = ones => D[m,n] = acc_m + acc_{m+16}.
  v2f a; a[0] = acc;  a[1] = 0.0f;
  v2f b; b[0] = 1.0f; b[1] = 1.0f;
  v8f c = {};
  v8f d = __builtin_amdgcn_wmma_f32_16x16x4_f32(
      /*neg_a=*/false, a, /*neg_b=*/false, b,
      /*c_mod=*/(short)0, c, /*reuse_a=*/false, /*reuse_b=*/false);
  // Lanes 0-15 see D rows 0..7, lanes 16-31 see rows 8..15.
  float s = d[0] + d[1] + d[2] + d[3] + d[4] + d[5] + d[6] + d[7];
  s += __shfl_xor(s, 16, 32);        // full 32-lane sum, all lanes

  __shared__ float wsum[NWAVE];
  const int lane = threadIdx.x & 31;
  const int wid  = threadIdx.x >> 5;
  if (lane == 0) wsum[wid] = s;
  __syncthreads();
  if (threadIdx.x == 0) {
    float t = 0.0f;
    for (int i = 0; i < NWAVE; ++i) t += wsum[i];  // fixed order: deterministic
    partials[blockIdx.x] = t;
  }
}

__global__ __launch_bounds__(256) void connect_loss_final(
    const float* __restrict__ partials, float* __restrict__ out, int n) {
  __shared__ double sd[256];
  double a = 0.0;
  for (int i = threadIdx.x; i < n; i += 256) a += (double)partials[i];
  sd[threadIdx.x] = a;
  __syncthreads();
  for (int off = 128; off > 0; off >>= 1) {
    if (threadIdx.x < (unsigned)off) sd[threadIdx.x] += sd[threadIdx.x + off];
    __syncthreads();
  }
  if (threadIdx.x == 0) out[0] = (float)sd[0];
}

extern "C" void kernel_launch(void* const* d_in, const int* in_sizes, int n_in,
                              void* d_out, int out_size, void* d_ws, size_t ws_size,
                              hipStream_t stream) {
  const float* x    = (const float*)d_in[0];
  const float* mask = (const float*)d_in[1];
  float* out = (float*)d_out;
  float* ws  = (float*)d_ws;   // needs NBLK * 4 = 8 KB
  const int n = in_sizes[0];   // 65536 * 512
  const int nsegs = n >> 4;    // 16-element segments
  connect_loss_partial<<<NBLK, NTHR, 0, stream>>>(x, mask, ws, nsegs);
  connect_loss_final<<<1, 256, 0, stream>>>(ws, out, NBLK);
}